// FlexInvariantPointAttention_40527311405661
// MI455X (gfx1250) — compile-verified
//
#include <hip/hip_runtime.h>
#include <math.h>

typedef __attribute__((ext_vector_type(2))) float v2f;
typedef __attribute__((ext_vector_type(8))) float v8f;

#define HH   12
#define NNN  512
#define CH_  16
#define PQK_ 4
#define PV_  8
#define INF_ 100000.0f
#define EPS_ 1e-8f

// ---------------------------------------------------------------------------
// Generic batched fp32 WMMA GEMM:  C[b] = A[b] @ W[b] (+ bias)
// One wave computes one 16x16 tile of C via V_WMMA_F32_16X16X4_F32.
// A layout (16x4, MxK): a[v] = A[m][k0 + 2*half + v]   (m = lane&15, half = lane>>4)
// B layout (4x16, KxN): b[v] = W[k0 + 2*half + v][n]   (n = lane&15)
// C layout (16x16):     acc[v] -> row (v + 8*half), col (lane&15)
// Ragged tiles: indices are CLAMPED in-bounds (no masked loads, no EXEC
// divergence); padded A-rows / B-cols only pollute padded C rows/cols,
// which are never stored.
// ---------------------------------------------------------------------------
__global__ __launch_bounds__(128) void gemm_wmma(
    const float* __restrict__ A, long lda, long strideA,
    const float* __restrict__ W, long ldw, long strideW,
    const float* __restrict__ bias,
    float* __restrict__ C, long ldc, long strideC,
    int M, int K, int Nc, int tiles_m, int tiles_n)
{
  int lane = threadIdx.x & 31;
  int wave = threadIdx.x >> 5;
  int tile = blockIdx.x * 4 + wave;
  if (tile >= tiles_m * tiles_n) return;          // wave-uniform exit: EXEC stays all-1s
  int tm = tile / tiles_n;
  int tn = tile % tiles_n;
  int b  = blockIdx.y;
  A += (size_t)b * strideA;
  W += (size_t)b * strideW;
  C += (size_t)b * strideC;

  int mloc = lane & 15;
  int half = lane >> 4;
  int row  = tm * 16 + mloc;                // A-operand row
  int col  = tn * 16 + mloc;                // B/C column
  bool cok = col < Nc;
  int rowc = (row < M)  ? row : (M - 1);    // clamped, in-bounds
  int colc = cok        ? col : (Nc - 1);

  const float* Ap = A + (size_t)rowc * lda + 2 * half;              // +=4 per step
  const float* Wp = W + (size_t)(2 * half) * ldw + colc;            // +=4*ldw per step

  v8f acc = {};
  #pragma unroll 4
  for (int k0 = 0; k0 < K; k0 += 4) {
    v2f av = *(const v2f*)Ap;               // global_load_b64 (8B aligned: lda even)
    v2f bv;
    bv.x = Wp[0];
    bv.y = Wp[ldw];
    acc = __builtin_amdgcn_wmma_f32_16x16x4_f32(false, av, false, bv,
                                                (short)0, acc, false, false);
    Ap += 4;
    Wp += 4 * (size_t)ldw;
  }
  float bb = (bias != nullptr && cok) ? bias[col] : 0.f;
  #pragma unroll
  for (int v = 0; v < 8; ++v) {
    int r = tm * 16 + v + 8 * half;
    if (r < M && cok) C[(size_t)r * ldc + col] = acc[v] + bb;
  }
}

// ---------------------------------------------------------------------------
// Apply per-residue frame (rotation + translation) to locally-projected points.
// pl: [N, 3, H*npts] (GEMM output).  First nsplit points/head -> outA, rest -> outB.
// outA: [N, H, nsplit, 3], outB: [N, H, npts-nsplit, 3]
// ---------------------------------------------------------------------------
__global__ void make_pts(const float* __restrict__ pl, const float* __restrict__ rot,
                         const float* __restrict__ trans,
                         float* __restrict__ outA, float* __restrict__ outB,
                         int npts, int nsplit)
{
  int idx = blockIdx.x * blockDim.x + threadIdx.x;
  int tot = NNN * HH * npts;
  if (idx >= tot) return;
  int n = idx / (HH * npts);
  int j = idx % (HH * npts);     // h*npts + p
  int h = j / npts, p = j % npts;
  int stride = 3 * HH * npts;
  const float* r = rot + n * 9;
  float lx = pl[(size_t)n * stride + 0 * HH * npts + j];
  float ly = pl[(size_t)n * stride + 1 * HH * npts + j];
  float lz = pl[(size_t)n * stride + 2 * HH * npts + j];
  float ox = r[0]*lx + r[1]*ly + r[2]*lz + trans[n*3+0];
  float oy = r[3]*lx + r[4]*ly + r[5]*lz + trans[n*3+1];
  float oz = r[6]*lx + r[7]*ly + r[8]*lz + trans[n*3+2];
  if (p < nsplit) {
    float* o = outA + ((size_t)n * HH * nsplit + h * nsplit + p) * 3;
    o[0] = ox; o[1] = oy; o[2] = oz;
  } else {
    int pv = p - nsplit, nb = npts - nsplit;
    float* o = outB + ((size_t)n * HH * nb + h * nb + pv) * 3;
    o[0] = ox; o[1] = oy; o[2] = oz;
  }
}

// ---------------------------------------------------------------------------
// Attention logits: S[h,i,j] = (q.k)/sqrt(48) + sqrt(1/3)*b[i,j,h]
//                   - 0.5*hw_h*(|qp|^2 + |kp|^2 - 2 qp.kp) + INF*(m_i m_j - 1)
// One wave per (h, 16x16 tile): 4 WMMAs (q.k, K=16) + 3 WMMAs (qp.kp, K=12).
// ---------------------------------------------------------------------------
__global__ __launch_bounds__(128) void ipa_scores(
    const float* __restrict__ q,     // [N, H*16]
    const float* __restrict__ kv,    // [N, H*32]  (k at +0, v at +16 per head)
    const float* __restrict__ qpts,  // [N, H*4*3]
    const float* __restrict__ kpts,  // [N, H*4*3]
    const float* __restrict__ bbias, // [N*N, 12]
    const float* __restrict__ mask,  // [N]
    const float* __restrict__ head_w,
    float* __restrict__ S)           // [H, N, N]
{
  int lane = threadIdx.x & 31;
  int wave = threadIdx.x >> 5;
  int wid  = blockIdx.x * 4 + wave;   // 0..1023
  int it = wid >> 5, jt = wid & 31;
  int h  = blockIdx.y;
  int i0 = it * 16, j0 = jt * 16;
  int mloc = lane & 15, half = lane >> 4;

  const float s1  = rsqrtf(48.f);          // 1/sqrt(3*C_H)
  const float s13 = sqrtf(1.f / 3.f);
  float hw = logf(1.f + expf(head_w[h])) * rsqrtf(54.f);  // softplus * sqrt(1/(3*18))

  // A operands: Q (K=16), Qpts (K=12) -- all 8B-aligned vector loads
  v2f aq[4], ap[3];
  const float* qrow = q + (size_t)(i0 + mloc) * (HH * CH_) + h * CH_ + 2 * half;
  #pragma unroll
  for (int s = 0; s < 4; ++s) aq[s] = *(const v2f*)(qrow + 4 * s);
  const float* qprow = qpts + (size_t)(i0 + mloc) * (HH * PQK_ * 3) + h * (PQK_ * 3) + 2 * half;
  float qp2 = 0.f;
  #pragma unroll
  for (int s = 0; s < 3; ++s) {
    ap[s] = *(const v2f*)(qprow + 4 * s);
    qp2 += ap[s].x * ap[s].x + ap[s].y * ap[s].y;
  }
  qp2 += __shfl_xor(qp2, 16, 32);          // |q_pts|^2 for row mloc
  float mi_r = mask[i0 + mloc];

  // B operands: K-tile (transposed access), Kpts-tile
  v2f bk[4], bp[3];
  const float* krow = kv + (size_t)(j0 + mloc) * (HH * 2 * CH_) + h * (2 * CH_) + 2 * half;
  #pragma unroll
  for (int s = 0; s < 4; ++s) bk[s] = *(const v2f*)(krow + 4 * s);
  const float* kprow = kpts + (size_t)(j0 + mloc) * (HH * PQK_ * 3) + h * (PQK_ * 3) + 2 * half;
  float kp2 = 0.f;
  #pragma unroll
  for (int s = 0; s < 3; ++s) {
    bp[s] = *(const v2f*)(kprow + 4 * s);
    kp2 += bp[s].x * bp[s].x + bp[s].y * bp[s].y;
  }
  kp2 += __shfl_xor(kp2, 16, 32);          // |k_pts|^2 for col mloc
  float mj = mask[j0 + mloc];

  v8f sacc = {}, dacc = {};
  #pragma unroll
  for (int s = 0; s < 4; ++s)
    sacc = __builtin_amdgcn_wmma_f32_16x16x4_f32(false, aq[s], false, bk[s], (short)0, sacc, false, false);
  #pragma unroll
  for (int s = 0; s < 3; ++s)
    dacc = __builtin_amdgcn_wmma_f32_16x16x4_f32(false, ap[s], false, bp[s], (short)0, dacc, false, false);

  #pragma unroll
  for (int v = 0; v < 8; ++v) {
    int row   = i0 + v + 8 * half;
    float qv  = __shfl(qp2,  v + 8 * half, 32);
    float miv = __shfl(mi_r, v + 8 * half, 32);
    float bb  = bbias[((size_t)row * NNN + (j0 + mloc)) * HH + h];
    float sc  = sacc[v] * s1 + s13 * bb + hw * dacc[v]
              - 0.5f * hw * (qv + kp2) + INF_ * (miv * mj - 1.f);
    S[(size_t)h * NNN * NNN + (size_t)row * NNN + (j0 + mloc)] = sc;
  }
}

// One wave per row of 512; in-place softmax.
__global__ __launch_bounds__(128) void softmax_rows(float* __restrict__ S)
{
  int lane = threadIdx.x & 31;
  int wave = threadIdx.x >> 5;
  int row  = blockIdx.x * 4 + wave;       // 0 .. H*N-1
  float* p = S + (size_t)row * NNN;
  float x[16];
  float m = -3.4e38f;
  #pragma unroll
  for (int t = 0; t < 16; ++t) { x[t] = p[lane + 32*t]; m = fmaxf(m, x[t]); }
  #pragma unroll
  for (int o = 16; o >= 1; o >>= 1) m = fmaxf(m, __shfl_xor(m, o, 32));
  float sum = 0.f;
  #pragma unroll
  for (int t = 0; t < 16; ++t) { x[t] = expf(x[t] - m); sum += x[t]; }
  #pragma unroll
  for (int o = 16; o >= 1; o >>= 1) sum += __shfl_xor(sum, o, 32);
  float inv = 1.f / sum;
  #pragma unroll
  for (int t = 0; t < 16; ++t) p[lane + 32*t] = x[t] * inv;
}

// Inverse-frame rotation of attended points + norms, scattered into feats.
__global__ void pts_post(const float* __restrict__ optraw,  // [N, H, 8, 3]
                         const float* __restrict__ rot, const float* __restrict__ trans,
                         float* __restrict__ feats)
{
  int idx = blockIdx.x * blockDim.x + threadIdx.x;
  if (idx >= NNN * HH * PV_) return;
  int n  = idx / (HH * PV_);
  int hp = idx % (HH * PV_);
  const float* o = optraw + ((size_t)n * HH * PV_ + hp) * 3;
  const float* r = rot + n * 9;
  float dx = o[0] - trans[n*3+0], dy = o[1] - trans[n*3+1], dz = o[2] - trans[n*3+2];
  float lx = r[0]*dx + r[3]*dy + r[6]*dz;     // R^T (o - t)
  float ly = r[1]*dx + r[4]*dy + r[7]*dz;
  float lz = r[2]*dx + r[5]*dy + r[8]*dz;
  float d  = sqrtf(lx*lx + ly*ly + lz*lz + EPS_);
  float* f = feats + (size_t)n * 960;
  f[192 + hp] = lx;
  f[288 + hp] = ly;
  f[384 + hp] = lz;
  f[480 + hp] = d;
}

extern "C" void kernel_launch(void* const* d_in, const int* in_sizes, int n_in,
                              void* d_out, int out_size, void* d_ws, size_t ws_size,
                              hipStream_t stream)
{
  const float* s     = (const float*)d_in[0];
  const float* z     = (const float*)d_in[1];
  const float* rot   = (const float*)d_in[2];
  const float* trans = (const float*)d_in[3];
  const float* mask  = (const float*)d_in[4];
  const float* Wq    = (const float*)d_in[5];
  const float* bq    = (const float*)d_in[6];
  const float* Wkv   = (const float*)d_in[7];
  const float* bkv   = (const float*)d_in[8];
  const float* Wqp   = (const float*)d_in[9];
  const float* bqp   = (const float*)d_in[10];
  const float* Wkvp  = (const float*)d_in[11];
  const float* bkvp  = (const float*)d_in[12];
  const float* Wb    = (const float*)d_in[13];
  const float* bbv   = (const float*)d_in[14];
  const float* Wdz   = (const float*)d_in[15];
  const float* bdz   = (const float*)d_in[16];
  const float* hwv   = (const float*)d_in[17];
  const float* Wout  = (const float*)d_in[18];
  const float* bout  = (const float*)d_in[19];
  float* out = (float*)d_out;

  float* ws     = (float*)d_ws;
  float* qbuf   = ws;  ws += 512*192;                 // q:      [N, H*16]
  float* kvbuf  = ws;  ws += 512*384;                 // kv:     [N, H*32]
  float* qploc  = ws;  ws += 512*144;                 // s@Wqp:  [N, 3*H*4]
  float* kvploc = ws;  ws += 512*432;                 // s@Wkvp: [N, 3*H*12]
  float* qpts   = ws;  ws += 512*144;                 // [N, H, 4, 3]
  float* kpts   = ws;  ws += 512*144;                 // [N, H, 4, 3]
  float* vpts   = ws;  ws += 512*288;                 // [N, H, 8, 3]
  float* bbias  = ws;  ws += (size_t)512*512*12;      // z@Wb:  [N*N, 12]
  float* pairz  = ws;  ws += (size_t)512*512*32;      // z@Wdz: [N*N, 32]
  float* S      = ws;  ws += (size_t)12*512*512;      // logits -> probs [H, N, N]
  float* optraw = ws;  ws += 512*288;                 // P@Vpts [N, H, 8, 3]
  float* feats  = ws;  ws += 512*960;                 // concat features

  auto gemm = [&](const float* A, long lda, long sA,
                  const float* W, long ldw, long sW, const float* bias,
                  float* C, long ldc, long sC, int M, int K, int Nc, int batch) {
    int tm = (M + 15) / 16, tn = (Nc + 15) / 16;
    dim3 grid((tm * tn + 3) / 4, batch);
    hipLaunchKernelGGL(gemm_wmma, grid, dim3(128), 0, stream,
                       A, lda, sA, W, ldw, sW, bias, C, ldc, sC, M, K, Nc, tm, tn);
  };

  // --- projections from s, and the two z-projections (z fits in 192MB L2) ---
  gemm(s, 384, 0, Wq,   192, 0, bq,   qbuf,   192, 0, 512, 384, 192, 1);
  gemm(s, 384, 0, Wkv,  384, 0, bkv,  kvbuf,  384, 0, 512, 384, 384, 1);
  gemm(s, 384, 0, Wqp,  144, 0, bqp,  qploc,  144, 0, 512, 384, 144, 1);
  gemm(s, 384, 0, Wkvp, 432, 0, bkvp, kvploc, 432, 0, 512, 384, 432, 1);
  gemm(z, 128, 0, Wb,    12, 0, bbv,  bbias,   12, 0, 512*512, 128, 12, 1);
  gemm(z, 128, 0, Wdz,   32, 0, bdz,  pairz,   32, 0, 512*512, 128, 32, 1);

  // --- frame transforms for points ---
  hipLaunchKernelGGL(make_pts, dim3((512*12*4  + 255)/256), dim3(256), 0, stream,
                     qploc,  rot, trans, qpts, qpts, 4, 4);
  hipLaunchKernelGGL(make_pts, dim3((512*12*12 + 255)/256), dim3(256), 0, stream,
                     kvploc, rot, trans, kpts, vpts, 12, 4);

  // --- attention logits (WMMA) + softmax ---
  hipLaunchKernelGGL(ipa_scores, dim3(256, 12), dim3(128), 0, stream,
                     qbuf, kvbuf, qpts, kpts, bbias, mask, hwv, S);
  hipLaunchKernelGGL(softmax_rows, dim3(12*512/4), dim3(128), 0, stream, S);

  // --- weighted sums, all as batched WMMA GEMMs ---
  // o = P @ V   -> feats[:, h*16 + c]      (batch over heads)
  gemm(S, 512, 512L*512, kvbuf + 16, 384, 32, nullptr, feats, 960, 16, 512, 512, 16, 12);
  // o_pt = P @ Vpts -> optraw[N,H,8,3]     (batch over heads)
  gemm(S, 512, 512L*512, vpts, 288, 24, nullptr, optraw, 288, 24, 512, 512, 24, 12);
  // o_pair[i,h,c] = sum_j P[h,i,j] pair_z[i,j,c]  (batch over query i)
  gemm(S, 512L*512, 512, pairz, 32, 512L*32, nullptr, feats + 576, 32, 960, 12, 512, 32, 512);

  // --- inverse rotation + point norms into feats[:, 192..576) ---
  hipLaunchKernelGGL(pts_post, dim3((512*96 + 255)/256), dim3(256), 0, stream,
                     optraw, rot, trans, feats);

  // --- output projection ---
  gemm(feats, 960, 0, Wout, 384, 0, bout, out, 384, 0, 512, 960, 384, 1);
}